// scaled_dot_product_attention_17635135717658
// MI455X (gfx1250) — compile-verified
//
#include <hip/hip_runtime.h>
#include <stdint.h>

// Scaled dot-product attention for MI455X (gfx1250), wave32 + fp32 WMMA +
// async global->LDS tile staging (ASYNCcnt double-buffering).
// Q,K,V: [B=2, H=8, S=2048, D=64] fp32. attn_mask: [B,H,S,S] bool (1 byte).
// Outputs (concatenated): context [B,H,S,64] fp32, then attn [B,H,S,S] fp32.

typedef float v2f __attribute__((ext_vector_type(2)));
typedef float v8f __attribute__((ext_vector_type(8)));

#define S_LEN     2048
#define DK        64
#define SC_STRIDE 2052   // 2048 + 4 pad -> row-major strip, rows land in distinct banks
#define KT_STRIDE 68     // 64 + 4 pad   -> same for the 16x64 tiles
#define NEG_BIG   (-1e9f)

#define TILE_FLOATS (16 * KT_STRIDE)
#define TILE_BYTES  (TILE_FLOATS * 4)

// LDS layout (floats): scores[16][2052] | tiles[2][16][68]
#define SMEM_FLOATS (16 * SC_STRIDE + 2 * TILE_FLOATS)

// ---- CDNA5 async global->LDS (ASYNCcnt) via inline asm (portable across
// toolchains per bridge doc). LDS byte offset is the low dword of the flat
// shared address (the aperture tag occupies the high dword).
__device__ __forceinline__ void async_ld_b128(uint32_t lds_byte_off, const float* g) {
    asm volatile("global_load_async_to_lds_b128 %0, %1, off"
                 :: "v"(lds_byte_off), "v"((uint64_t)(uintptr_t)g)
                 : "memory");
}
#define WAIT_ASYNC(n) asm volatile("s_wait_asynccnt " #n ::: "memory")

// Stage one contiguous 16x64 fp32 tile (4KB) into a padded 16x68 LDS buffer:
// 8 async B128s, 512B per instruction, fully coalesced, no VGPR round trip.
__device__ __forceinline__ void stage_tile_async(uint32_t buf_byte_base,
                                                 const float* __restrict__ src,
                                                 int lane) {
    #pragma unroll
    for (int i = 0; i < 8; ++i) {
        int ci  = i * 32 + lane;                 // 16-byte chunk id, 0..255
        int row = ci >> 4, col = (ci & 15) * 4;  // never crosses a row
        async_ld_b128(buf_byte_base + (uint32_t)(row * KT_STRIDE + col) * 4u,
                      src + ci * 4);
    }
}

__global__ __launch_bounds__(32) void sdpa_gfx1250_kernel(
    const float* __restrict__ Q, const float* __restrict__ K,
    const float* __restrict__ V, const uint8_t* __restrict__ mask,
    float* __restrict__ ctx_out, float* __restrict__ attn_out)
{
    extern __shared__ float smem[];
    float* sc    = smem;                     // 16 x SC_STRIDE score/probability strip
    float* tiles = smem + 16 * SC_STRIDE;    // 2 consecutive 16x68 tile buffers
    // Single base + arithmetic offset (NOT a pointer array) so addrspace(3)
    // inference survives and the compiler emits ds_load_* with imm offsets.
    const uint32_t tiles_off = (uint32_t)(uintptr_t)tiles;

    const int lane = threadIdx.x & 31;
    const int r16  = lane & 15;             // M (or N) index within 16
    const int hi   = lane >> 4;             // selects K-halves / M-halves per WMMA layout

    const int bh   = blockIdx.x >> 7;       // 0..15  (b*8+h)
    const int q0   = (blockIdx.x & 127) * 16;

    const float*   Qb = Q    + (size_t)bh * S_LEN * DK;
    const float*   Kb = K    + (size_t)bh * S_LEN * DK;
    const float*   Vb = V    + (size_t)bh * S_LEN * DK;
    const uint8_t* Mb = mask + (size_t)bh * S_LEN * S_LEN;
    float*         Cb = ctx_out  + (size_t)bh * S_LEN * DK;
    float*         Ab = attn_out + (size_t)bh * S_LEN * S_LEN;

    // ---- Stage Q block (16x64 contiguous) via async, extract A-fragments ----
    stage_tile_async(tiles_off + TILE_BYTES, Qb + (size_t)q0 * DK, lane);
    WAIT_ASYNC(0);
    // a[s][v] = Q[q0 + r16][4s + 2*hi + v] * (1/sqrt(64))  (scale folded into Q)
    v2f aq[16];
    #pragma unroll
    for (int s = 0; s < 16; ++s) {
        v2f t = *(const v2f*)(tiles + TILE_FLOATS + r16 * KT_STRIDE + 4 * s + 2 * hi);
        aq[s].x = t.x * 0.125f;
        aq[s].y = t.y * 0.125f;
    }

    // ---- Phase 1: masked scores, async double-buffered K tiles ----
    stage_tile_async(tiles_off, Kb, lane);            // prologue: tile 0 -> buf 0
    #pragma unroll 2
    for (int kt = 0; kt < 128; ++kt) {
        const int k0 = kt * 16;
        if (kt + 1 < 128) {
            stage_tile_async(tiles_off + ((kt + 1) & 1) * TILE_BYTES,
                             Kb + (size_t)(k0 + 16) * DK, lane);
            // prefetch next tile's mask bytes (global_prefetch_b8)
            __builtin_prefetch(Mb + (size_t)(q0 + r16) * S_LEN + k0 + 16, 0, 1);
            WAIT_ASYNC(8);   // in-order completion: current tile's 8 loads landed
        } else {
            WAIT_ASYNC(0);
        }
        const float* kp = tiles + (kt & 1) * TILE_FLOATS;

        // S_tile = (Q*scale) @ K_tile^T : 16 chained fp32 WMMAs over D=64
        v8f c = {};
        #pragma unroll
        for (int s = 0; s < 16; ++s) {
            v2f b = *(const v2f*)(kp + r16 * KT_STRIDE + 4 * s + 2 * hi);
            c = __builtin_amdgcn_wmma_f32_16x16x4_f32(
                    false, aq[s], false, b, (short)0, c, false, false);
        }

        // Apply bool mask (true -> -1e9); C layout: c[v] = S[q0+v+8hi][k0+r16]
        #pragma unroll
        for (int v = 0; v < 8; ++v) {
            int qrow = v + 8 * hi;
            uint8_t m = Mb[(size_t)(q0 + qrow) * S_LEN + k0 + r16];
            sc[qrow * SC_STRIDE + k0 + r16] = m ? NEG_BIG : c[v];
        }
    }

    // ---- Softmax over each 2048-long row (2 lanes/row, float4-vectorized) ----
    {
        float* rp = sc + r16 * SC_STRIDE;
        float mrow = -3.4e38f;
        for (int j = hi * 4; j < S_LEN; j += 8) {
            float4 x = *(const float4*)(rp + j);
            mrow = fmaxf(mrow, fmaxf(fmaxf(x.x, x.y), fmaxf(x.z, x.w)));
        }
        mrow = fmaxf(mrow, __shfl_xor(mrow, 16, 32));

        float sum = 0.0f;
        for (int j = hi * 4; j < S_LEN; j += 8) {
            float4 x = *(float4*)(rp + j);
            x.x = __expf(x.x - mrow);
            x.y = __expf(x.y - mrow);
            x.z = __expf(x.z - mrow);
            x.w = __expf(x.w - mrow);
            *(float4*)(rp + j) = x;          // unnormalized exp back to LDS
            sum += (x.x + x.y) + (x.z + x.w);
        }
        sum += __shfl_xor(sum, 16, 32);
        float inv = 1.0f / sum;

        // Normalize + stream attn to HBM: 512B per wave iteration (b128 stores).
        for (int rr = 0; rr < 16; ++rr) {
            float iv   = __shfl(inv, rr, 32);    // lanes rr and rr+16 hold row rr's inv
            float* rrp = sc + rr * SC_STRIDE;
            float* gap = Ab + (size_t)(q0 + rr) * S_LEN;
            for (int j = lane * 4; j < S_LEN; j += 128) {
                float4 p = *(const float4*)(rrp + j);
                p.x *= iv; p.y *= iv; p.z *= iv; p.w *= iv;
                *(float4*)(rrp + j) = p;     // normalized P stays in LDS for phase 2
                *(float4*)(gap + j) = p;
            }
        }
    }

    // ---- Phase 2: context = P @ V, async double-buffered V tiles ----
    v8f acc0 = {}, acc1 = {}, acc2 = {}, acc3 = {};
    stage_tile_async(tiles_off, Vb, lane);            // prologue: V rows 0..15 -> buf 0
    #pragma unroll 2
    for (int vt = 0; vt < 128; ++vt) {
        const int v0 = vt * 16;
        if (vt + 1 < 128) {
            stage_tile_async(tiles_off + ((vt + 1) & 1) * TILE_BYTES,
                             Vb + (size_t)(v0 + 16) * DK, lane);
            WAIT_ASYNC(8);
        } else {
            WAIT_ASYNC(0);
        }
        const float* vb = tiles + (vt & 1) * TILE_FLOATS;

        #pragma unroll
        for (int kl = 0; kl < 16; kl += 4) {
            // A-fragment from probabilities: a[v] = P[q0+r16][v0 + kl + 2hi + v]
            v2f a = *(const v2f*)(sc + r16 * SC_STRIDE + v0 + kl + 2 * hi);
            // B-fragments from staged V tile: b[v] = V[v0+kl+2hi+v][n0 + r16]
            const float* row0 = vb + (kl + 2 * hi) * KT_STRIDE + r16;
            const float* row1 = row0 + KT_STRIDE;
            v2f b0 = { row0[0],  row1[0]  };
            v2f b1 = { row0[16], row1[16] };
            v2f b2 = { row0[32], row1[32] };
            v2f b3 = { row0[48], row1[48] };

            acc0 = __builtin_amdgcn_wmma_f32_16x16x4_f32(false, a, false, b0, (short)0, acc0, false, false);
            acc1 = __builtin_amdgcn_wmma_f32_16x16x4_f32(false, a, false, b1, (short)0, acc1, false, false);
            acc2 = __builtin_amdgcn_wmma_f32_16x16x4_f32(false, a, false, b2, (short)0, acc2, false, false);
            acc3 = __builtin_amdgcn_wmma_f32_16x16x4_f32(false, a, false, b3, (short)0, acc3, false, false);
        }
    }

    // Context store: D[q0 + v + 8hi][n]; lanes 0-15 write contiguous 64B runs.
    #pragma unroll
    for (int v = 0; v < 8; ++v) {
        float* crow = Cb + (size_t)(q0 + v + 8 * hi) * DK + r16;
        crow[0]  = acc0[v];
        crow[16] = acc1[v];
        crow[32] = acc2[v];
        crow[48] = acc3[v];
    }
}

extern "C" void kernel_launch(void* const* d_in, const int* in_sizes, int n_in,
                              void* d_out, int out_size, void* d_ws, size_t ws_size,
                              hipStream_t stream) {
    (void)in_sizes; (void)n_in; (void)d_ws; (void)ws_size; (void)out_size;

    const float*   Q    = (const float*)d_in[0];
    const float*   K    = (const float*)d_in[1];
    const float*   V    = (const float*)d_in[2];
    const uint8_t* mask = (const uint8_t*)d_in[3];   // jnp.bool_ -> 1 byte/elem

    float* ctx  = (float*)d_out;                                   // [2,8,2048,64]
    float* attn = (float*)d_out + (size_t)2 * 8 * S_LEN * DK;      // [2,8,2048,2048]

    const size_t smem_bytes = SMEM_FLOATS * sizeof(float);         // ~137 KB (<320 KB/WGP)
    static_assert(SMEM_FLOATS * sizeof(float) < 320 * 1024, "LDS budget");

    hipFuncSetAttribute((const void*)sdpa_gfx1250_kernel,
                        hipFuncAttributeMaxDynamicSharedMemorySize, (int)smem_bytes);

    dim3 grid(16 * (S_LEN / 16));   // 2048 one-wave workgroups (bh * qblock)
    dim3 block(32);                 // one wave32
    sdpa_gfx1250_kernel<<<grid, block, smem_bytes, stream>>>(Q, K, V, mask, ctx, attn);
}